// Model_90280212562334
// MI455X (gfx1250) — compile-verified
//
#include <hip/hip_runtime.h>
#include <hip/hip_bf16.h>
#include <math.h>
#include <stdint.h>

#define BB   16
#define TT   3000
#define NN   128
#define WINL 30
#define WN   100
#define BWC  (BB*WN)     // 1600 windows total
#define DD   64
#define HID  128
#define KC   32
#define G4   (4*HID)     // 512

typedef float v2f __attribute__((ext_vector_type(2)));
typedef float v8f __attribute__((ext_vector_type(8)));

// gfx1250 async global->LDS copy (16B per lane), tracked by ASYNCcnt
__device__ __forceinline__ void async_ld_b128(const float* lds_dst, const float* gsrc) {
    unsigned ldsa = (unsigned)(uintptr_t)(const void*)lds_dst;   // low 32 bits = LDS byte offset
    unsigned long long ga = (unsigned long long)(uintptr_t)(const void*)gsrc;
    asm volatile("global_load_async_to_lds_b128 %0, %1, off"
                 :: "v"(ldsa), "v"(ga) : "memory");
}
__device__ __forceinline__ void async_wait0() {
    asm volatile("s_wait_asynccnt 0" ::: "memory");
}

// ---------------- zero init ----------------
__global__ void k_zero(float* p, int n) {
    int i = blockIdx.x * blockDim.x + threadIdx.x;
    if (i < n) p[i] = 0.f;
}

// ---------------- SignalRep: conv1 -> relu -> conv2(WMMA implicit GEMM) -> relu -> mean -> proj -> S ----------------
// one wave per sequence, 8 sequences per 256-thread block
__global__ __launch_bounds__(256) void k_signal(
    const float* __restrict__ x,
    const float* __restrict__ w1, const float* __restrict__ b1,
    const float* __restrict__ w2, const float* __restrict__ b2,
    const float* __restrict__ pw, const float* __restrict__ pb,
    const float* __restrict__ proto,
    float* __restrict__ Hs, float* __restrict__ S)
{
    __shared__ float seq[8][WINL];
    __shared__ float h1p[8][32][36];    // conv1 out, zero-padded [ci][t+2]; reused as out2 [cout][t]
    __shared__ float w2t[5*32*32];      // transposed conv2 weights [k][ci][cout]
    __shared__ float w1s[32*5];
    __shared__ float b1s[32], b2s[32];
    __shared__ float ms[8][32];
    __shared__ float hsb[8][64];

    int tid = threadIdx.x;
    long s0 = (long)blockIdx.x * 8;

    for (int i = tid; i < 5*32*32; i += 256) {
        int kf = i >> 10, ci = (i >> 5) & 31, co = i & 31;
        w2t[i] = w2[(co*32 + ci)*5 + kf];
    }
    for (int i = tid; i < 32*5; i += 256) w1s[i] = w1[i];
    if (tid < 32) { b1s[tid] = b1[tid]; b2s[tid] = b2[tid]; }
    for (int i = tid; i < 8*WINL; i += 256) {
        int q = i & 7, l = i >> 3;
        long s = s0 + q;
        long n = s % NN, bw = s / NN, w = bw % WN, b = bw / WN;
        seq[q][l] = x[(b*TT + w*WINL + l)*NN + n];
    }
    __syncthreads();

    int q = tid >> 5;      // sequence within block
    int c = tid & 31;      // lane / channel
    // conv1 (1->32, k=5, pad 2) + relu, into padded [t+2] layout (zeros at tt {0,1,32..35})
    for (int tt = 0; tt < 36; ++tt) {
        float v = 0.f;
        int t = tt - 2;
        if (t >= 0 && t < WINL) {
            float acc = b1s[c];
            #pragma unroll
            for (int k = 0; k < 5; ++k) {
                int p = t + k - 2;
                float xv = (p >= 0 && p < WINL) ? seq[q][p] : 0.f;
                acc += w1s[c*5 + k] * xv;
            }
            v = fmaxf(acc, 0.f);
        }
        h1p[q][c][tt] = v;
    }
    __syncthreads();

    // conv2 as 5 shifted 32x32x32 GEMMs: out2[c_out][t] = sum_k sum_ci W2k[c_out][ci] * h1p[ci][t+k]
    const int m  = c & 15;
    const int kb = (c >> 4) * 2;
    const int rh = 8 * (c >> 4);
    const float* hb = &h1p[q][0][0];
    v8f ct[4];
    #pragma unroll
    for (int ti = 0; ti < 2; ++ti) {
        #pragma unroll
        for (int tj = 0; tj < 2; ++tj) {
            v8f cc = {0.f,0.f,0.f,0.f,0.f,0.f,0.f,0.f};
            #pragma unroll
            for (int kf = 0; kf < 5; ++kf) {
                const float* wk = &w2t[kf*1024];
                for (int kk = 0; kk < 32; kk += 4) {
                    v2f a, bf;
                    a.x  = wk[(kk+kb+0)*32 + ti*16 + m];
                    a.y  = wk[(kk+kb+1)*32 + ti*16 + m];
                    bf.x = hb[(kk+kb+0)*36 + tj*16 + m + kf];
                    bf.y = hb[(kk+kb+1)*36 + tj*16 + m + kf];
                    cc = __builtin_amdgcn_wmma_f32_16x16x4_f32(false, a, false, bf, (short)0, cc, false, false);
                }
            }
            ct[ti*2+tj] = cc;
        }
    }
    // relu(out2 + b2) written back over h1p[q] as [cout][t] (t=30,31 garbage but never read)
    #pragma unroll
    for (int ti = 0; ti < 2; ++ti)
        #pragma unroll
        for (int tj = 0; tj < 2; ++tj)
            #pragma unroll
            for (int v = 0; v < 8; ++v) {
                int row = ti*16 + v + rh;
                int col = tj*16 + m;
                h1p[q][row][col] = fmaxf(ct[ti*2+tj][v] + b2s[row], 0.f);
            }
    __syncthreads();
    // mean over t<30
    float mm = 0.f;
    for (int t = 0; t < WINL; ++t) mm += h1p[q][c][t];
    ms[q][c] = mm * (1.f/WINL);
    __syncthreads();
    // proj (32->64): lane c produces j=c and j=c+32
    long s = s0 + q;
    #pragma unroll
    for (int jj = 0; jj < 2; ++jj) {
        int j = c + jj*32;
        float acc = pb[j];
        for (int ci = 0; ci < 32; ++ci) acc += ms[q][ci] * pw[ci*64 + j];
        hsb[q][j] = acc;
        Hs[s*64 + j] = acc;
    }
    __syncthreads();
    // S = softmax_k(Hs . proto[k]) ; lane = k
    float lg = 0.f;
    for (int j = 0; j < 64; ++j) lg += hsb[q][j] * proto[c*64 + j];
    float mx = lg;
    #pragma unroll
    for (int o = 16; o > 0; o >>= 1) mx = fmaxf(mx, __shfl_xor(mx, o, 32));
    float e = expf(lg - mx);
    float sm = e;
    #pragma unroll
    for (int o = 16; o > 0; o >>= 1) sm += __shfl_xor(sm, o, 32);
    S[s*32 + c] = e / sm;
}

// ---------------- per-window: corr/A, AS=A@S, Ac=S^T AS, Xc=S^T Hs, An normalize (WMMA f32 16x16x4) ----------------
// dyn LDS (floats): Af[16384] | xm/AS[4096] | Ssh[4096] | Hss[8192]  = 32768 floats
__global__ __launch_bounds__(256) void k_pool(
    const float* __restrict__ x,
    const float* __restrict__ Hsg, const float* __restrict__ Sg,
    float* __restrict__ Xc, float* __restrict__ An)
{
    extern __shared__ float smem[];
    float* Af  = smem;                 // 16384
    float* xmm = smem + 16384;         // 4096  (k-major xmm[k*128+n]); aliased as AS later
    float* Ssh = smem + 20480;         // 4096  (Ssh[n*32+k])
    float* Hss = smem + 24576;         // 8192  (Hss[n*64+f])
    float* AS  = xmm;                  // AS[n*32+l]
    __shared__ float rstd[NN];
    __shared__ float Acs[KC][KC];
    __shared__ float Asy[KC][KC];
    __shared__ float dinv[KC];

    int tid  = threadIdx.x;
    int lane = tid & 31, wv = tid >> 5;
    int m  = lane & 15;
    int kb = (lane >> 4) * 2;
    int rh = 8 * (lane >> 4);
    long bw = blockIdx.x;
    long b = bw / WN, w = bw % WN;

    // async staging: x window rows 0..29 (3840 contiguous floats), S tile, Hs tile
    const float* xsrc = x + (b*TT + w*WINL)*NN;
    const float* Ssrc = Sg  + bw*(NN*KC);
    const float* Hsrc = Hsg + bw*(NN*DD);
    for (int i = tid; i < (WINL*NN)/4; i += 256) async_ld_b128(xmm + i*4, xsrc + i*4);
    for (int i = tid; i < (NN*KC)/4;  i += 256) async_ld_b128(Ssh + i*4, Ssrc + i*4);
    for (int i = tid; i < (NN*DD)/4;  i += 256) async_ld_b128(Hss + i*4, Hsrc + i*4);
    xmm[WINL*NN + tid] = 0.f;          // zero pad rows 30,31 (256 floats)
    async_wait0();
    __syncthreads();

    // column mean-subtract + rstd
    if (tid < NN) {
        int n = tid;
        float mu = 0.f;
        for (int l = 0; l < WINL; ++l) mu += xmm[l*128 + n];
        mu *= (1.f/WINL);
        float ss = 0.f;
        for (int l = 0; l < WINL; ++l) { float v = xmm[l*128 + n] - mu; xmm[l*128 + n] = v; ss += v*v; }
        float var = fmaxf(ss / (WINL - 1 + 1e-8f), 1e-8f);
        rstd[n] = rsqrtf(var);
    }
    __syncthreads();

    const float inv29 = 1.f / (WINL - 1 + 1e-8f);
    // cov tiles: wave wv owns row-tile ti=wv
    {
        int ti = wv;
        for (int tj = 0; tj < 8; ++tj) {
            v8f c = {0.f,0.f,0.f,0.f,0.f,0.f,0.f,0.f};
            for (int kk = 0; kk < 32; kk += 4) {
                v2f a, bf;
                a.x  = xmm[(kk+kb+0)*128 + ti*16 + m];
                a.y  = xmm[(kk+kb+1)*128 + ti*16 + m];
                bf.x = xmm[(kk+kb+0)*128 + tj*16 + m];
                bf.y = xmm[(kk+kb+1)*128 + tj*16 + m];
                c = __builtin_amdgcn_wmma_f32_16x16x4_f32(false, a, false, bf, (short)0, c, false, false);
            }
            #pragma unroll
            for (int v = 0; v < 8; ++v) {
                int r  = ti*16 + v + rh;
                int cn = tj*16 + m;
                float cv = c[v] * inv29 * rstd[r] * rstd[cn];
                if (!__builtin_isfinite(cv)) cv = 0.f;
                Af[r*128 + cn] = (r == cn) ? 1.f : cv;   // A = corr, diag forced to 1
            }
        }
    }
    __syncthreads();   // Af done, xmm reads done (AS alias now safe)

    // AS = Af @ S : wave wv -> row-tile wv, col tiles 0..1 (K=128)
    {
        int ti = wv;
        for (int tj = 0; tj < 2; ++tj) {
            v8f c = {0.f,0.f,0.f,0.f,0.f,0.f,0.f,0.f};
            for (int kk = 0; kk < 128; kk += 4) {
                v2f a, bf;
                a.x  = Af[(ti*16 + m)*128 + kk+kb+0];
                a.y  = Af[(ti*16 + m)*128 + kk+kb+1];
                bf.x = Ssh[(kk+kb+0)*32 + tj*16 + m];
                bf.y = Ssh[(kk+kb+1)*32 + tj*16 + m];
                c = __builtin_amdgcn_wmma_f32_16x16x4_f32(false, a, false, bf, (short)0, c, false, false);
            }
            #pragma unroll
            for (int v = 0; v < 8; ++v)
                AS[(ti*16 + v + rh)*32 + tj*16 + m] = c[v];
        }
    }
    __syncthreads();

    // 12 tiles: t<4 -> Ac = S^T @ AS (2x2) ; t>=4 -> Xc = S^T @ Hs (2x4) ; K=128
    for (int t = wv; t < 12; t += 8) {
        int isAc = (t < 4);
        int ti = isAc ? (t >> 1) : ((t-4) >> 2);
        int tj = isAc ? (t & 1)  : ((t-4) & 3);
        v8f c = {0.f,0.f,0.f,0.f,0.f,0.f,0.f,0.f};
        for (int kk = 0; kk < 128; kk += 4) {
            v2f a, bf;
            a.x = Ssh[(kk+kb+0)*32 + ti*16 + m];   // (S^T)[row][k] = S[k][row]
            a.y = Ssh[(kk+kb+1)*32 + ti*16 + m];
            if (isAc) {
                bf.x = AS[(kk+kb+0)*32 + tj*16 + m];
                bf.y = AS[(kk+kb+1)*32 + tj*16 + m];
            } else {
                bf.x = Hss[(kk+kb+0)*64 + tj*16 + m];
                bf.y = Hss[(kk+kb+1)*64 + tj*16 + m];
            }
            c = __builtin_amdgcn_wmma_f32_16x16x4_f32(false, a, false, bf, (short)0, c, false, false);
        }
        #pragma unroll
        for (int v = 0; v < 8; ++v) {
            int r  = ti*16 + v + rh;
            int cn = tj*16 + m;
            if (isAc) Acs[r][cn] = c[v];
            else      Xc[(bw*KC + r)*DD + cn] = c[v];
        }
    }
    __syncthreads();

    // An = D^-1/2 (sym(Ac)+I) D^-1/2
    if (tid < KC) {
        int r = tid; float rs = 0.f;
        for (int cn = 0; cn < KC; ++cn) {
            float v = 0.5f*(Acs[r][cn] + Acs[cn][r]) + ((r == cn) ? 1.f : 0.f);
            Asy[r][cn] = v; rs += v;
        }
        dinv[r] = rsqrtf(fmaxf(rs, 1e-8f));
    }
    __syncthreads();
    for (int i = tid; i < KC*KC; i += 256) {
        int r = i >> 5, cn = i & 31;
        An[bw*KC*KC + i] = dinv[r] * Asy[r][cn] * dinv[cn];
    }
}

// ---------------- LSTM step: P = X@Wx + Hp@Wh + (bx+bh), k-blocked (8 rows/thread, X/Hp staged in LDS) ----------------
// grid: 16b * 4kq * 2colhalf = 128 blocks of 256
__global__ __launch_bounds__(256) void k_gates(
    const float* __restrict__ Xc,
    const float* __restrict__ Wx, const float* __restrict__ bx,
    const float* __restrict__ Wh, const float* __restrict__ bh,
    const float* __restrict__ H1, const float* __restrict__ H2,
    float* __restrict__ P, float coef, float dnv, int t)
{
    __shared__ float xs[8][64];
    __shared__ float hp[8][128];
    int tid = threadIdx.x;
    int bid = blockIdx.x;
    int ch = bid & 1, kq = (bid >> 1) & 3, b = bid >> 3;
    int k0 = kq * 8;
    for (int i = tid; i < 8*64; i += 256) {
        int kk = i >> 6, j = i & 63;
        xs[kk][j] = Xc[(((long)b*WN + t)*KC + k0 + kk)*DD + j];
    }
    for (int i = tid; i < 8*128; i += 256) {
        int kk = i >> 7, m2 = i & 127;
        long hi = ((long)b*KC + k0 + kk)*HID + m2;
        hp[kk][m2] = (coef*H1[hi] + H2[hi]) * dnv;
    }
    __syncthreads();
    int col = ch*256 + tid;
    float bias = bx[col] + bh[col];
    float acc[8];
    #pragma unroll
    for (int kk = 0; kk < 8; ++kk) acc[kk] = bias;

    for (int j4 = 0; j4 < DD; j4 += 4) {
        float4 xv[8];
        #pragma unroll
        for (int kk = 0; kk < 8; ++kk) xv[kk] = *(const float4*)&xs[kk][j4];
        #pragma unroll
        for (int dj = 0; dj < 4; ++dj) {
            float wv = Wx[(j4+dj)*G4 + col];
            #pragma unroll
            for (int kk = 0; kk < 8; ++kk) {
                float xe = (dj==0) ? xv[kk].x : (dj==1) ? xv[kk].y : (dj==2) ? xv[kk].z : xv[kk].w;
                acc[kk] += xe * wv;
            }
        }
    }
    for (int m4 = 0; m4 < HID; m4 += 4) {
        float4 hv[8];
        #pragma unroll
        for (int kk = 0; kk < 8; ++kk) hv[kk] = *(const float4*)&hp[kk][m4];
        #pragma unroll
        for (int dj = 0; dj < 4; ++dj) {
            float wv = Wh[(m4+dj)*G4 + col];
            #pragma unroll
            for (int kk = 0; kk < 8; ++kk) {
                float he = (dj==0) ? hv[kk].x : (dj==1) ? hv[kk].y : (dj==2) ? hv[kk].z : hv[kk].w;
                acc[kk] += he * wv;
            }
        }
    }
    #pragma unroll
    for (int kk = 0; kk < 8; ++kk)
        P[((long)b*KC + k0 + kk)*G4 + col] = acc[kk];
}

// ---------------- LSTM step: g = An@P, gate math, write H/C, accumulate cluster-mean ----------------
__global__ __launch_bounds__(256) void k_lstm(
    const float* __restrict__ An, const float* __restrict__ P,
    float* __restrict__ C, float* __restrict__ Hout,
    float* __restrict__ Hbar, int t)
{
    int i = blockIdx.x * blockDim.x + threadIdx.x;       // BB*KC*HID
    if (i >= BB*KC*HID) return;
    int h = i & 127;
    int k = (i >> 7) & 31;
    int b = i >> 12;
    const float* ar  = &An[(((long)b*WN + t)*KC + k)*KC];
    const float* pbb = &P[(long)b*KC*G4];
    float gi = 0.f, gf = 0.f, go = 0.f, gc = 0.f;
    for (int m2 = 0; m2 < KC; ++m2) {
        float a = ar[m2];
        const float* pr = pbb + m2*G4;
        gi += a * pr[h];
        gf += a * pr[h + 128];
        go += a * pr[h + 256];
        gc += a * pr[h + 384];
    }
    float sgi = 1.f/(1.f + expf(-gi));
    float sgf = 1.f/(1.f + expf(-gf));
    float sgo = 1.f/(1.f + expf(-go));
    long ci = ((long)b*KC + k)*HID + h;
    float Cn = sgf*C[ci] + sgi*tanhf(gc);
    float Hn = sgo*tanhf(Cn);
    C[ci] = Cn;
    Hout[ci] = Hn;
    atomicAdd(&Hbar[((long)b*WN + t)*HID + h], Hn * (1.f/KC));
}

// ---------------- mean over nodes of Hs ----------------
__global__ void k_nodemean(const float* __restrict__ Hs, float* __restrict__ Hsbar)
{
    int i = blockIdx.x * blockDim.x + threadIdx.x;       // BWC*DD
    if (i >= BWC*DD) return;
    int f = i & 63; long bw = i >> 6;
    float s = 0.f;
    const float* p = &Hs[bw*NN*DD + f];
    for (int n = 0; n < NN; ++n) s += p[n*DD];
    Hsbar[i] = s * (1.f/NN);
}

// ---------------- temporal attention pooling + classifier ----------------
__global__ __launch_bounds__(128) void k_final(
    const float* __restrict__ Hbar, const float* __restrict__ Hsbar,
    const float* __restrict__ qt, const float* __restrict__ qs,
    const float* __restrict__ cw, const float* __restrict__ cb,
    float* __restrict__ out)
{
    __shared__ float red[128];
    __shared__ float dots[WN];
    __shared__ float zg[HID], zs[DD];
    int b = blockIdx.x, tid = threadIdx.x;

    red[tid] = (tid < HID) ? qt[tid]*qt[tid] : 0.f;
    __syncthreads();
    for (int o = 64; o > 0; o >>= 1) { if (tid < o) red[tid] += red[tid + o]; __syncthreads(); }
    float qtn = sqrtf(red[0]) + 1e-8f;
    __syncthreads();
    if (tid < WN) {
        const float* Z = &Hbar[((long)b*WN + tid)*HID];
        float d = 0.f;
        for (int h = 0; h < HID; ++h) d += Z[h]*qt[h];
        dots[tid] = d / qtn;
    }
    __syncthreads();
    if (tid == 0) {
        float mx = -1e30f;
        for (int t2 = 0; t2 < WN; ++t2) mx = fmaxf(mx, dots[t2]);
        float sm = 0.f;
        for (int t2 = 0; t2 < WN; ++t2) { float e = expf(dots[t2]-mx); dots[t2] = e; sm += e; }
        float inv = 1.f/sm;
        for (int t2 = 0; t2 < WN; ++t2) dots[t2] *= inv;
    }
    __syncthreads();
    if (tid < HID) {
        float s = 0.f;
        for (int t2 = 0; t2 < WN; ++t2) s += dots[t2]*Hbar[((long)b*WN + t2)*HID + tid];
        zg[tid] = s;
    }
    __syncthreads();
    red[tid] = (tid < DD) ? qs[tid]*qs[tid] : 0.f;
    __syncthreads();
    for (int o = 64; o > 0; o >>= 1) { if (tid < o) red[tid] += red[tid + o]; __syncthreads(); }
    float qsn = sqrtf(red[0]) + 1e-8f;
    __syncthreads();
    if (tid < WN) {
        const float* Z = &Hsbar[((long)b*WN + tid)*DD];
        float d = 0.f;
        for (int f = 0; f < DD; ++f) d += Z[f]*qs[f];
        dots[tid] = d / qsn;
    }
    __syncthreads();
    if (tid == 0) {
        float mx = -1e30f;
        for (int t2 = 0; t2 < WN; ++t2) mx = fmaxf(mx, dots[t2]);
        float sm = 0.f;
        for (int t2 = 0; t2 < WN; ++t2) { float e = expf(dots[t2]-mx); dots[t2] = e; sm += e; }
        float inv = 1.f/sm;
        for (int t2 = 0; t2 < WN; ++t2) dots[t2] *= inv;
    }
    __syncthreads();
    if (tid < DD) {
        float s = 0.f;
        for (int t2 = 0; t2 < WN; ++t2) s += dots[t2]*Hsbar[((long)b*WN + t2)*DD + tid];
        zs[tid] = s;
    }
    __syncthreads();
    float p = 0.f;
    if (tid < HID) p  = zg[tid]*cw[tid];
    if (tid < DD)  p += zs[tid]*cw[HID + tid];
    red[tid] = p;
    __syncthreads();
    for (int o = 64; o > 0; o >>= 1) { if (tid < o) red[tid] += red[tid + o]; __syncthreads(); }
    if (tid == 0) out[b] = 1.f/(1.f + expf(-(red[0] + cb[0])));
}

// ---------------- workspace layout (floats) ----------------
#define OFF_HS    0L
#define OFF_S     13107200L
#define OFF_XC    19660800L
#define OFF_AN    22937600L
#define OFF_HBAR  24576000L
#define OFF_HA    24780800L
#define OFF_HB    24846336L
#define OFF_C     24911872L
#define OFF_HSBAR 24977408L
#define OFF_P     25079808L
// total 25341952 floats = ~101.4 MB

extern "C" void kernel_launch(void* const* d_in, const int* in_sizes, int n_in,
                              void* d_out, int out_size, void* d_ws, size_t ws_size,
                              hipStream_t stream) {
    const float* x_enc  = (const float*)d_in[0];
    const float* conv1w = (const float*)d_in[1];
    const float* conv1b = (const float*)d_in[2];
    const float* conv2w = (const float*)d_in[3];
    const float* conv2b = (const float*)d_in[4];
    const float* projw  = (const float*)d_in[5];
    const float* projb  = (const float*)d_in[6];
    const float* proto  = (const float*)d_in[7];
    const float* Wx     = (const float*)d_in[8];
    const float* bx     = (const float*)d_in[9];
    const float* Wh     = (const float*)d_in[10];
    const float* bh     = (const float*)d_in[11];
    const float* q_t    = (const float*)d_in[12];
    const float* q_s    = (const float*)d_in[13];
    const float* cls_w  = (const float*)d_in[14];
    const float* cls_b  = (const float*)d_in[15];
    float* out = (float*)d_out;
    float* ws  = (float*)d_ws;

    float* Hs    = ws + OFF_HS;
    float* S     = ws + OFF_S;
    float* Xc    = ws + OFF_XC;
    float* An    = ws + OFF_AN;
    float* Hbar  = ws + OFF_HBAR;
    float* Ha    = ws + OFF_HA;
    float* Hb    = ws + OFF_HB;
    float* C     = ws + OFF_C;
    float* Hsbar = ws + OFF_HSBAR;
    float* P     = ws + OFF_P;

    // zero Hbar + Ha + Hb + C (contiguous 401408 floats)
    {
        int n = 401408;
        k_zero<<<(n + 255)/256, 256, 0, stream>>>(Hbar, n);
    }
    // SignalRep: 204800 sequences, 8 per block (conv2 via WMMA)
    k_signal<<<(BB*WN*NN)/8, 256, 0, stream>>>(x_enc, conv1w, conv1b, conv2w, conv2b,
                                               projw, projb, proto, Hs, S);
    // mean over nodes for z_signal pooling
    {
        int n = BWC*DD;
        k_nodemean<<<(n + 255)/256, 256, 0, stream>>>(Hs, Hsbar);
    }
    // per-window correlation + graph pooling (WMMA + async-to-LDS staging)
    k_pool<<<BWC, 256, 32768*sizeof(float), stream>>>(x_enc, Hs, S, Xc, An);
    // GCLSTM recurrence: 100 sequential steps, ping-pong H buffers
    float* Hc = Ha;   // H_{t-1}
    float* Ho = Hb;   // H_{t-2}; receives H_t
    for (int t = 0; t < WN; ++t) {
        float coef = (t == 0) ? 1.f : 2.f;
        float dnv  = (t == 0) ? 1.f : ((t == 1) ? 0.5f : (1.f/3.f));
        k_gates<<<128, 256, 0, stream>>>(Xc, Wx, bx, Wh, bh, Hc, Ho, P, coef, dnv, t);
        k_lstm<<<(BB*KC*HID + 255)/256, 256, 0, stream>>>(An, P, C, Ho, Hbar, t);
        float* tmp = Hc; Hc = Ho; Ho = tmp;
    }
    // attention pooling + classifier
    k_final<<<BB, 128, 0, stream>>>(Hbar, Hsbar, q_t, q_s, cls_w, cls_b, out);
}